// ModelWithLogCalibration_34093450396428
// MI455X (gfx1250) — compile-verified
//
#include <hip/hip_runtime.h>
#include <stdint.h>

#define ROWS    64
#define NCOL    524288
#define KTOP    128
#define OUT_F   128
#define SLICES  8          // streaming blocks per row -> 512 blocks total
#define TPB     512
#define NBUCK   2048       // top 11 bits of order-preserving key
#define CANDM   4096       // candidate cap per row

typedef __attribute__((ext_vector_type(2))) float v2f;
typedef __attribute__((ext_vector_type(8))) float v8f;

// Order-preserving map: descending float order == descending unsigned order.
__device__ __forceinline__ uint32_t map_key(float f) {
  uint32_t u = __float_as_uint(f);
  return (u & 0x80000000u) ? ~u : (u | 0x80000000u);
}
__device__ __forceinline__ float unmap_key(uint32_t k) {
  uint32_t u = (k & 0x80000000u) ? (k & 0x7FFFFFFFu) : ~k;
  return __uint_as_float(u);
}

// ---------------------------------------------------------------- init: zero ws
__global__ __launch_bounds__(TPB) void init_kernel(uint32_t* ghist, uint32_t* gcnt) {
  const int total = ROWS * NBUCK;
  for (int i = blockIdx.x * TPB + threadIdx.x; i < total; i += gridDim.x * TPB)
    ghist[i] = 0u;
  if (blockIdx.x == 0 && threadIdx.x < ROWS) gcnt[threadIdx.x] = 0u;
}

// ---------------------------------------------------- pass 1: bucket histogram
__global__ __launch_bounds__(TPB) void hist_kernel(const float* __restrict__ pred,
                                                   uint32_t* __restrict__ ghist) {
  __shared__ uint32_t hist[NBUCK];
  const int row   = blockIdx.x / SLICES;
  const int slice = blockIdx.x % SLICES;
  for (int i = threadIdx.x; i < NBUCK; i += TPB) hist[i] = 0u;
  __syncthreads();

  const int per = (NCOL / 4) / SLICES;   // float4 elements per slice
  const float4* __restrict__ p4 =
      reinterpret_cast<const float4*>(pred + (size_t)row * NCOL) + (size_t)slice * per;
  for (int i = threadIdx.x; i < per; i += TPB) {
    if (i + 4 * TPB < per) __builtin_prefetch(&p4[i + 4 * TPB], 0, 0);
    float4 v = p4[i];
    atomicAdd(&hist[map_key(v.x) >> 21], 1u);
    atomicAdd(&hist[map_key(v.y) >> 21], 1u);
    atomicAdd(&hist[map_key(v.z) >> 21], 1u);
    atomicAdd(&hist[map_key(v.w) >> 21], 1u);
  }
  __syncthreads();

  uint32_t* gh = ghist + (size_t)row * NBUCK;
  for (int i = threadIdx.x; i < NBUCK; i += TPB)
    if (hist[i]) atomicAdd(&gh[i], hist[i]);
}

// ------------------------------------- select: suffix-scan -> threshold bucket
__global__ __launch_bounds__(256) void select_kernel(const uint32_t* __restrict__ ghist,
                                                     uint32_t* __restrict__ tb_out) {
  const int row = blockIdx.x;
  const uint32_t* gh = ghist + (size_t)row * NBUCK;
  __shared__ uint32_t chunk[256];   // chunk c = sum of 8 buckets, c=0 is the top
  const int c = threadIdx.x;
  uint32_t s = 0;
  const int hi = NBUCK - 8 * c;     // buckets [hi-8, hi)
  for (int b = hi - 1; b >= hi - 8; --b) s += gh[b];
  chunk[c] = s;
  __syncthreads();
  if (threadIdx.x == 0) {
    uint32_t cum = 0, tb = 0;
    for (int cc = 0; cc < 256; ++cc) {
      if (cum + chunk[cc] >= KTOP) {            // threshold is inside chunk cc
        const int top = NBUCK - 8 * cc;
        for (int b = top - 1; b >= top - 8; --b) {
          cum += gh[b];
          if (cum >= KTOP) { tb = (uint32_t)b; break; }
        }
        break;
      }
      cum += chunk[cc];
    }
    tb_out[row] = tb;
  }
}

// ---------------------------- pass 2 (L2-resident): compact candidates >= tb
__global__ __launch_bounds__(TPB) void compact_kernel(const float* __restrict__ pred,
                                                      const uint32_t* __restrict__ tb_in,
                                                      uint32_t* __restrict__ gcnt,
                                                      unsigned long long* __restrict__ gcand) {
  const int row   = blockIdx.x / SLICES;
  const int slice = blockIdx.x % SLICES;
  const uint32_t tb = tb_in[row];
  const int per = (NCOL / 4) / SLICES;
  const float4* __restrict__ p4 =
      reinterpret_cast<const float4*>(pred + (size_t)row * NCOL) + (size_t)slice * per;
  const int base0 = slice * per * 4;
  unsigned long long* cr = gcand + (size_t)row * CANDM;

  for (int i = threadIdx.x; i < per; i += TPB) {
    float4 v = p4[i];
    const int base = base0 + i * 4;
    uint32_t k;
    // pack {key, ~index}: 64-bit descending sort => value desc, index asc on ties
    k = map_key(v.x);
    if ((k >> 21) >= tb) { uint32_t p = atomicAdd(&gcnt[row], 1u);
      if (p < CANDM) cr[p] = ((unsigned long long)k << 32) | (uint32_t)~(uint32_t)(base + 0); }
    k = map_key(v.y);
    if ((k >> 21) >= tb) { uint32_t p = atomicAdd(&gcnt[row], 1u);
      if (p < CANDM) cr[p] = ((unsigned long long)k << 32) | (uint32_t)~(uint32_t)(base + 1); }
    k = map_key(v.z);
    if ((k >> 21) >= tb) { uint32_t p = atomicAdd(&gcnt[row], 1u);
      if (p < CANDM) cr[p] = ((unsigned long long)k << 32) | (uint32_t)~(uint32_t)(base + 2); }
    k = map_key(v.w);
    if ((k >> 21) >= tb) { uint32_t p = atomicAdd(&gcnt[row], 1u);
      if (p < CANDM) cr[p] = ((unsigned long long)k << 32) | (uint32_t)~(uint32_t)(base + 3); }
  }
}

// --------------------- bitonic sort candidates in LDS, emit topv + indices
__global__ __launch_bounds__(TPB) void sort_kernel(const uint32_t* __restrict__ gcnt,
                                                   const unsigned long long* __restrict__ gcand,
                                                   float* __restrict__ topv,
                                                   int* __restrict__ idx_out) {
  __shared__ unsigned long long cand[CANDM];
  const int row = blockIdx.x;
  uint32_t cnt = gcnt[row]; if (cnt > CANDM) cnt = CANDM;
  const unsigned long long* cr = gcand + (size_t)row * CANDM;
  for (int i = threadIdx.x; i < CANDM; i += TPB)
    cand[i] = (i < (int)cnt) ? cr[i] : 0ull;
  __syncthreads();

  for (int k = 2; k <= CANDM; k <<= 1) {
    for (int j = k >> 1; j > 0; j >>= 1) {
      for (int i = threadIdx.x; i < CANDM; i += TPB) {
        const int ij = i ^ j;
        if (ij > i) {
          unsigned long long a = cand[i], b = cand[ij];
          const bool desc = ((i & k) == 0);
          if (desc ? (a < b) : (a > b)) { cand[i] = b; cand[ij] = a; }
        }
      }
      __syncthreads();
    }
  }

  if (threadIdx.x < KTOP) {
    const unsigned long long p = cand[threadIdx.x];
    topv[row * KTOP + threadIdx.x]    = unmap_key((uint32_t)(p >> 32));
    idx_out[row * KTOP + threadIdx.x] = (int)~(uint32_t)p;
  }
}

// --------------- sigmoid(topv @ W^T + b) via V_WMMA_F32_16X16X4_F32 (wave32)
// y[m][n] = sum_k A[m][k] * W[n][k];  B[k][n] := W[n][k]
__global__ __launch_bounds__(256) void linear_sigmoid_wmma_kernel(
    const float* __restrict__ topv, const float* __restrict__ W,
    const float* __restrict__ bias, float* __restrict__ out) {
  const int lane  = threadIdx.x & 31;
  const int ntile = threadIdx.x >> 5;   // 8 waves -> 8 N-tiles
  const int mtile = blockIdx.x;         // 4 blocks -> 4 M-tiles
  const int half  = lane >> 4;          // A/B 16x4 layout: lanes 16..31 hold K+{2,3}
  const int l16   = lane & 15;

  const float* __restrict__ arow = topv + (size_t)(mtile * 16 + l16) * OUT_F; // A[m][k]
  const float* __restrict__ brow = W    + (size_t)(ntile * 16 + l16) * OUT_F; // B[k][n]=W[n][k]

  v8f acc = {};
#pragma unroll
  for (int k0 = 0; k0 < OUT_F; k0 += 4) {
    v2f a, b;
    a.x = arow[k0 + 2 * half + 0];
    a.y = arow[k0 + 2 * half + 1];
    b.x = brow[k0 + 2 * half + 0];
    b.y = brow[k0 + 2 * half + 1];
    acc = __builtin_amdgcn_wmma_f32_16x16x4_f32(
        /*neg_a=*/false, a, /*neg_b=*/false, b,
        /*c_mod=*/(short)0, acc, /*reuse_a=*/false, /*reuse_b=*/false);
  }

  const int n  = ntile * 16 + l16;
  const float bn = bias[n];
#pragma unroll
  for (int r = 0; r < 8; ++r) {                 // C/D layout: M = r + 8*half
    const int m = mtile * 16 + r + 8 * half;
    const float x = acc[r] + bn;
    out[m * OUT_F + n] = 1.0f / (1.0f + expf(-x));
  }
}

extern "C" void kernel_launch(void* const* d_in, const int* in_sizes, int n_in,
                              void* d_out, int out_size, void* d_ws, size_t ws_size,
                              hipStream_t stream) {
  const float* pred = (const float*)d_in[0];   // [64, 524288]
  const float* W    = (const float*)d_in[1];   // [128, 128]
  const float* bias = (const float*)d_in[2];   // [128]

  float* out  = (float*)d_out;                       // outputs [64,128] f32
  int*   idxs = (int*)d_out + ROWS * KTOP;           // idxs    [64,128] i32

  // workspace carve-up (all re-initialized every call; graph-replay safe)
  char* ws = (char*)d_ws;
  uint32_t*           ghist = (uint32_t*)(ws);                               // 512 KiB
  uint32_t*           gcnt  = (uint32_t*)(ws + (size_t)ROWS * NBUCK * 4);    // 256 B
  uint32_t*           tb    = gcnt + ROWS;                                   // 256 B
  unsigned long long* gcand = (unsigned long long*)(ws + 528384);            // 2 MiB
  float*              topv  = (float*)(ws + 528384 + (size_t)ROWS * CANDM * 8); // 32 KiB

  init_kernel   <<<64,           TPB, 0, stream>>>(ghist, gcnt);
  hist_kernel   <<<ROWS*SLICES,  TPB, 0, stream>>>(pred, ghist);
  select_kernel <<<ROWS,         256, 0, stream>>>(ghist, tb);
  compact_kernel<<<ROWS*SLICES,  TPB, 0, stream>>>(pred, tb, gcnt, gcand);
  sort_kernel   <<<ROWS,         TPB, 0, stream>>>(gcnt, gcand, topv, idxs);
  linear_sigmoid_wmma_kernel<<<ROWS/16, 256, 0, stream>>>(topv, W, bias, out);
}